// GaussianSplat3D_24764781429331
// MI455X (gfx1250) — compile-verified
//
#include <hip/hip_runtime.h>
#include <math.h>
#include <stdint.h>

#define N_G 1024
#define IMG_W 256
#define IMG_H 256

typedef __attribute__((ext_vector_type(2))) float v2f;
typedef __attribute__((ext_vector_type(8))) float v8f;
typedef __attribute__((ext_vector_type(4))) unsigned int v4u;
typedef __attribute__((ext_vector_type(4))) int v4i;
typedef __attribute__((ext_vector_type(8))) int v8i;

// ---------------------------------------------------------------------------
// Kernel 1: preprocess all 1024 gaussians, bitonic-sort by depth in LDS,
// write sorted 12-float records (px,py,ia,ib | ic,opEff,r,g | b,0,0,0) to ws.
// rec[9] == 0.0 doubles as the "zero color" slot for B-matrix columns >= 3.
// ---------------------------------------------------------------------------
__global__ __launch_bounds__(1024)
void gs_prep_sort(const float* __restrict__ means,
                  const float* __restrict__ log_scales,
                  const float* __restrict__ colors,
                  const float* __restrict__ opacities,
                  const float* __restrict__ intr,
                  const float* __restrict__ c2w,
                  float* __restrict__ rec_out)
{
    __shared__ float skey[N_G];
    __shared__ int   sidx[N_G];
    __shared__ float sp[N_G][9];

    const int i = threadIdx.x;

    float mx = means[3*i+0], my = means[3*i+1], mz = means[3*i+2];
    float s0 = __expf(log_scales[3*i+0]);
    float s1 = __expf(log_scales[3*i+1]);
    float s2 = __expf(log_scales[3*i+2]);
    float d0 = s0*s0 + 1e-6f, d1 = s1*s1 + 1e-6f, d2 = s2*s2 + 1e-6f;

    // W = R_w2c = R_c2w^T ; c2w is row-major 4x4
    float W00 = c2w[0], W01 = c2w[4], W02 = c2w[8];
    float W10 = c2w[1], W11 = c2w[5], W12 = c2w[9];
    float W20 = c2w[2], W21 = c2w[6], W22 = c2w[10];
    float tc0 = c2w[3], tc1 = c2w[7], tc2 = c2w[11];
    float tw0 = -(W00*tc0 + W01*tc1 + W02*tc2);
    float tw1 = -(W10*tc0 + W11*tc1 + W12*tc2);
    float tw2 = -(W20*tc0 + W21*tc1 + W22*tc2);

    float x = W00*mx + W01*my + W02*mz + tw0;
    float y = W10*mx + W11*my + W12*mz + tw1;
    float z = W20*mx + W21*my + W22*mz + tw2;

    float c00 = W00*W00*d0 + W01*W01*d1 + W02*W02*d2;
    float c01 = W00*W10*d0 + W01*W11*d1 + W02*W12*d2;
    float c02 = W00*W20*d0 + W01*W21*d1 + W02*W22*d2;
    float c11 = W10*W10*d0 + W11*W11*d1 + W12*W12*d2;
    float c12 = W10*W20*d0 + W11*W21*d1 + W12*W22*d2;
    float c22 = W20*W20*d0 + W21*W21*d1 + W22*W22*d2;

    float fx = intr[0], cx = intr[2], fy = intr[4], cy = intr[5];
    bool  vis = z > 1e-4f;
    float sz  = vis ? z : 1.0f;
    float inv = 1.0f / sz;
    float px = fx * x * inv + cx;
    float py = fy * y * inv + cy;

    float u0 = fx * inv;
    float u2 = -fx * x * inv * inv;
    float v1 = fy * inv;
    float v2 = -fy * y * inv * inv;

    float a2 = u0*u0*c00 + 2.0f*u0*u2*c02 + u2*u2*c22 + 1e-4f;
    float b2 = u0*v1*c01 + u0*v2*c02 + u2*v1*c12 + u2*v2*c22;
    float cc = v1*v1*c11 + 2.0f*v1*v2*c12 + v2*v2*c22 + 1e-4f;
    float det = a2*cc - b2*b2;
    float idet = 1.0f / det;
    float ia =  cc * idet;
    float ib = -b2 * idet;
    float ic =  a2 * idet;

    float op = 1.0f / (1.0f + __expf(-opacities[i]));
    float r  = fminf(fmaxf(colors[3*i+0], 0.0f), 1.0f);
    float g  = fminf(fmaxf(colors[3*i+1], 0.0f), 1.0f);
    float b  = fminf(fmaxf(colors[3*i+2], 0.0f), 1.0f);

    bool mask = vis && (px >= 0.0f) && (px < (float)IMG_W)
                    && (py >= 0.0f) && (py < (float)IMG_H);

    sp[i][0] = px; sp[i][1] = py; sp[i][2] = ia; sp[i][3] = ib;
    sp[i][4] = ic; sp[i][5] = mask ? op : 0.0f;
    sp[i][6] = r;  sp[i][7] = g;  sp[i][8] = b;
    skey[i] = mask ? z : INFINITY;
    sidx[i] = i;
    __syncthreads();

    for (int k = 2; k <= N_G; k <<= 1) {
        for (int j = k >> 1; j > 0; j >>= 1) {
            int ixj = i ^ j;
            if (ixj > i) {
                bool up = ((i & k) == 0);
                float ki = skey[i], kj = skey[ixj];
                if (up ? (ki > kj) : (ki < kj)) {
                    skey[i] = kj; skey[ixj] = ki;
                    int t = sidx[i]; sidx[i] = sidx[ixj]; sidx[ixj] = t;
                }
            }
            __syncthreads();
        }
    }

    int src = sidx[i];
    float4* ro = (float4*)rec_out;
    float4 r0 = make_float4(sp[src][0], sp[src][1], sp[src][2], sp[src][3]);
    float4 r1 = make_float4(sp[src][4], sp[src][5], sp[src][6], sp[src][7]);
    float4 r2 = make_float4(sp[src][8], 0.0f, 0.0f, 0.0f);
    ro[i*3+0] = r0; ro[i*3+1] = r1; ro[i*3+2] = r2;
}

// ---------------------------------------------------------------------------
// Kernel 2: each wave composites 32 distinct pixels front-to-back.
// Weights for 4 gaussians/lane are swapped across wave halves (shfl_xor 16)
// into the V_WMMA_F32_16X16X4_F32 A-layout, feeding TWO 16x16 f32 tiles
// (pixels 0-15 and 16-31). B columns 0..2 carry RGB via a branch-free LDS
// gather (column>=3 reads the record's zero pad). The 48KB sorted-record
// table is staged into LDS once per block by the Tensor Data Mover.
// ---------------------------------------------------------------------------
__global__ __launch_bounds__(256)
void gs_render(const float* __restrict__ rec, float* __restrict__ out)
{
    __shared__ float gsh[N_G * 12];   // 48 KB

    // ---- TDM: DMA global -> LDS, issued by wave 0 only ----
    if (threadIdx.x < 32) {
        unsigned lds_base = (unsigned)(size_t)&gsh[0];           // low 32 bits = LDS offset
        unsigned long long ga = (unsigned long long)(size_t)rec;
        // D# group0: count=1 | lds_addr | global_addr[56:0] | type=2
        v4u g0 = { 1u,
                   lds_base,
                   (unsigned)(ga & 0xFFFFFFFFu),
                   (unsigned)((ga >> 32) & 0x1FFFFFFu) | 0x80000000u };
        // D# group1: wg_mask=0, data_size=2(4B); tensor_dim0=12288, tensor_dim1=1,
        // tile_dim0=12288, tile_dim1=1, tensor_dim0_stride=12288 (1-D copy)
        v8i g1 = { (int)0x20000u,
                   (int)(12288u << 16),   // tensor_dim0[15:0] @ bits 63:48
                   (int)(1u << 16),       // tensor_dim0 hi | tensor_dim1 lo
                   (int)(12288u << 16),   // tensor_dim1 hi | tile_dim0
                   1,                     // tile_dim1=1 | tile_dim2=0
                   12288,                 // tensor_dim0_stride[31:0]
                   0, 0 };
        v4i gz4 = { 0, 0, 0, 0 };
        v8i gz8 = { 0, 0, 0, 0, 0, 0, 0, 0 };
        __builtin_amdgcn_tensor_load_to_lds(g0, g1, gz4, gz4, gz8, 0);
        __builtin_amdgcn_s_wait_tensorcnt(0);
    }
    __syncthreads();

    const int lane = threadIdx.x & 31;
    const int wave = threadIdx.x >> 5;
    const bool hi  = lane >= 16;
    const int  m   = lane & 15;                   // tile row / B,D column
    const int  wbase = (blockIdx.x * 8 + wave) * 32;
    const int  p   = wbase + lane;                // this lane's pixel
    const float xf = (float)(p & (IMG_W - 1));
    const float yf = (float)(p >> 8);
    const int chIdx = 6 + (m < 3 ? m : 3);        // r,g,b or zero pad

    float T = 1.0f;
    v8f acc0 = {0.f,0.f,0.f,0.f,0.f,0.f,0.f,0.f}; // pixels wbase+0..15
    v8f acc1 = {0.f,0.f,0.f,0.f,0.f,0.f,0.f,0.f}; // pixels wbase+16..31

    for (int gb = 0; gb < N_G; gb += 4) {
        float w0, w1, w2, w3;
        {
            const float* q = &gsh[(gb+0)*12];
            float dx = xf - q[0], dy = yf - q[1];
            float pw = -0.5f*(q[2]*dx*dx + q[4]*dy*dy + 2.0f*q[3]*dx*dy);
            float al = q[5] * __expf(pw);
            w0 = al * T; T *= (1.0f - al);
        }
        {
            const float* q = &gsh[(gb+1)*12];
            float dx = xf - q[0], dy = yf - q[1];
            float pw = -0.5f*(q[2]*dx*dx + q[4]*dy*dy + 2.0f*q[3]*dx*dy);
            float al = q[5] * __expf(pw);
            w1 = al * T; T *= (1.0f - al);
        }
        {
            const float* q = &gsh[(gb+2)*12];
            float dx = xf - q[0], dy = yf - q[1];
            float pw = -0.5f*(q[2]*dx*dx + q[4]*dy*dy + 2.0f*q[3]*dx*dy);
            float al = q[5] * __expf(pw);
            w2 = al * T; T *= (1.0f - al);
        }
        {
            const float* q = &gsh[(gb+3)*12];
            float dx = xf - q[0], dy = yf - q[1];
            float pw = -0.5f*(q[2]*dx*dx + q[4]*dy*dy + 2.0f*q[3]*dx*dy);
            float al = q[5] * __expf(pw);
            w3 = al * T; T *= (1.0f - al);
        }

        // B-matrix gather: vgpr0 = rows K0/K2, vgpr1 = rows K1/K3
        int kb = hi ? 2 : 0;
        v2f bv;
        bv.x = gsh[(gb + kb    ) * 12 + chIdx];
        bv.y = gsh[(gb + kb + 1) * 12 + chIdx];

        // Swap weights across wave halves to build both A tiles
        float w0x = __shfl_xor(w0, 16, 32);
        float w1x = __shfl_xor(w1, 16, 32);
        float w2x = __shfl_xor(w2, 16, 32);
        float w3x = __shfl_xor(w3, 16, 32);
        v2f a0; a0.x = hi ? w2x : w0;  a0.y = hi ? w3x : w1;  // rows = pixels 0-15
        v2f a1; a1.x = hi ? w2  : w0x; a1.y = hi ? w3  : w1x; // rows = pixels 16-31

        acc0 = __builtin_amdgcn_wmma_f32_16x16x4_f32(
                   false, a0, false, bv, (short)0, acc0, false, false);
        acc1 = __builtin_amdgcn_wmma_f32_16x16x4_f32(
                   false, a1, false, bv, (short)0, acc1, false, false);
    }

    // D layout: lane n, vgpr v -> (row = v + (hi?8:0), col = n&15)
    if (m < 3) {
        int p0 = wbase + (hi ? 8 : 0);
        int p1 = wbase + 16 + (hi ? 8 : 0);
#pragma unroll
        for (int v = 0; v < 8; ++v) {
            out[(p0 + v) * 3 + m] = acc0[v];
            out[(p1 + v) * 3 + m] = acc1[v];
        }
    }
}

extern "C" void kernel_launch(void* const* d_in, const int* in_sizes, int n_in,
                              void* d_out, int out_size, void* d_ws, size_t ws_size,
                              hipStream_t stream) {
    const float* means      = (const float*)d_in[0];
    const float* log_scales = (const float*)d_in[1];
    const float* colors     = (const float*)d_in[2];
    const float* opacities  = (const float*)d_in[3];
    const float* intrinsics = (const float*)d_in[4];
    const float* cam2world  = (const float*)d_in[5];

    float* rec = (float*)d_ws;   // 1024 * 12 floats = 48 KB sorted records
    gs_prep_sort<<<1, 1024, 0, stream>>>(means, log_scales, colors, opacities,
                                         intrinsics, cam2world, rec);
    gs_render<<<(IMG_H * IMG_W) / 256, 256, 0, stream>>>(rec, (float*)d_out);
}